// MessagePassingLayer_70918499992212
// MI455X (gfx1250) — compile-verified
//
#include <hip/hip_runtime.h>
#include <hip/hip_bf16.h>

typedef __attribute__((ext_vector_type(2))) float v2f;
typedef __attribute__((ext_vector_type(8))) float v8f;

#define IN_DIM 128
#define OUT_DIM 128

// ---------------------------------------------------------------------------
// Kernel 1: zero the workspace (agg[N*128] ++ deg[N]), contiguous floats.
// ---------------------------------------------------------------------------
__global__ __launch_bounds__(512) void mp_zero_kernel(float* __restrict__ p, long long n) {
    long long i = (long long)blockIdx.x * blockDim.x + threadIdx.x;
    if (i < n) p[i] = 0.0f;
}

// ---------------------------------------------------------------------------
// Kernel 2: edge scatter. One wave (32 lanes) per edge: lane reads float4 of
// x[src], atomically adds into agg[dst]; lane 0 counts degree.
// edge_index is int64, flat layout: src = ei[0..E), dst = ei[E..2E).
// ---------------------------------------------------------------------------
__global__ __launch_bounds__(256) void mp_scatter_kernel(const float* __restrict__ x,
                                                         const long long* __restrict__ ei,
                                                         float* __restrict__ agg,
                                                         float* __restrict__ deg,
                                                         long long E) {
    long long t = (long long)blockIdx.x * blockDim.x + threadIdx.x;
    long long e = t >> 5;
    if (e >= E) return;
    int lane = (int)(t & 31);

    long long src = ei[e];
    long long dst = ei[E + e];

    const float4 v = *(const float4*)(x + src * IN_DIM + lane * 4);
    float* ap = agg + dst * IN_DIM + lane * 4;
    unsafeAtomicAdd(ap + 0, v.x);   // -> global_atomic_add_f32 (L2-resident target)
    unsafeAtomicAdd(ap + 1, v.y);
    unsafeAtomicAdd(ap + 2, v.z);
    unsafeAtomicAdd(ap + 3, v.w);
    if (lane == 0) unsafeAtomicAdd(deg + dst, 1.0f);
}

// ---------------------------------------------------------------------------
// Kernel 3: fused dual-GEMM epilogue using V_WMMA_F32_16X16X4_F32.
//   out = relu( x @ Ws^T + (agg*rdeg) @ Wn^T + (bs+bn) )
// One wave computes one 16(M)x16(N) tile; 8 waves per block cover the 128
// output channels for one 16-node row tile. K-loop: 32 iters x 2 wmma.
//
// 32-bit A 16x4 layout: lanes 0-15 hold M=lane {K=k0,k0+1}; lanes 16-31 hold
// M=lane-16 {K=k0+2,k0+3}. B 4x16: lane = N column, vgpr = K row (same half
// split). C/D: VGPR r -> row M=r (lanes 0-15) / M=r+8 (lanes 16-31), lane&15=N.
// ---------------------------------------------------------------------------
__global__ __launch_bounds__(256) void mp_gemm_kernel(const float* __restrict__ x,
                                                      const float* __restrict__ agg,
                                                      const float* __restrict__ deg,
                                                      const float* __restrict__ Ws,
                                                      const float* __restrict__ Wn,
                                                      const float* __restrict__ bs,
                                                      const float* __restrict__ bn,
                                                      float* __restrict__ out,
                                                      int Nn) {
    const int wave = threadIdx.x >> 5;      // 0..7 -> N tile
    const int lane = threadIdx.x & 31;
    const int half = lane >> 4;             // 0: K+0,1  1: K+2,3
    const int l15  = lane & 15;

    const int mbase = blockIdx.x * 16;
    const int nbase = wave * 16;

    int arow = mbase + l15;
    if (arow >= Nn) arow = Nn - 1;          // clamp; EXEC stays all-ones for WMMA
    const int bcol = nbase + l15;           // output channel this lane owns in B/C/D

    const float rdeg = 1.0f / fmaxf(deg[arow], 1.0f);

    const float* __restrict__ xr = x   + (size_t)arow * IN_DIM;
    const float* __restrict__ ar = agg + (size_t)arow * IN_DIM;
    const float* __restrict__ ws = Ws  + (size_t)bcol * IN_DIM;  // W[n][k]
    const float* __restrict__ wn = Wn  + (size_t)bcol * IN_DIM;

    v8f acc = {};
    #pragma unroll
    for (int k0 = 0; k0 < IN_DIM; k0 += 4) {
        const int kb = k0 + 2 * half;

        v2f a_s; a_s.x = xr[kb];  a_s.y = xr[kb + 1];
        v2f b_s; b_s.x = ws[kb];  b_s.y = ws[kb + 1];
        acc = __builtin_amdgcn_wmma_f32_16x16x4_f32(false, a_s, false, b_s,
                                                    (short)0, acc, false, false);

        v2f a_n; a_n.x = ar[kb] * rdeg;  a_n.y = ar[kb + 1] * rdeg;
        v2f b_n; b_n.x = wn[kb];         b_n.y = wn[kb + 1];
        acc = __builtin_amdgcn_wmma_f32_16x16x4_f32(false, a_n, false, b_n,
                                                    (short)0, acc, false, false);
    }

    const float bias = bs[bcol] + bn[bcol];

    #pragma unroll
    for (int r = 0; r < 8; ++r) {
        const int row = mbase + r + 8 * half;
        if (row < Nn) {
            out[(size_t)row * OUT_DIM + bcol] = fmaxf(acc[r] + bias, 0.0f);
        }
    }
}

// ---------------------------------------------------------------------------
extern "C" void kernel_launch(void* const* d_in, const int* in_sizes, int n_in,
                              void* d_out, int out_size, void* d_ws, size_t ws_size,
                              hipStream_t stream) {
    const float*     x  = (const float*)d_in[0];
    const long long* ei = (const long long*)d_in[1];
    const float*     Ws = (const float*)d_in[2];
    const float*     bs = (const float*)d_in[3];
    const float*     Wn = (const float*)d_in[4];
    const float*     bn = (const float*)d_in[5];
    float*           out = (float*)d_out;

    const int       Nn = in_sizes[0] / IN_DIM;
    const long long E  = (long long)in_sizes[1] / 2;

    float* agg = (float*)d_ws;                       // N*128 f32
    float* deg = agg + (size_t)Nn * IN_DIM;          // N f32

    // 1) zero agg + deg (contiguous)
    const long long ztot = (long long)Nn * IN_DIM + Nn;
    mp_zero_kernel<<<(unsigned)((ztot + 511) / 512), 512, 0, stream>>>(agg, ztot);

    // 2) scatter-add: 32 threads per edge
    const long long sthreads = E * 32;
    mp_scatter_kernel<<<(unsigned)((sthreads + 255) / 256), 256, 0, stream>>>(
        x, ei, agg, deg, E);

    // 3) fused mean + dual WMMA GEMM + bias + relu
    const unsigned mtiles = (unsigned)((Nn + 15) / 16);
    mp_gemm_kernel<<<mtiles, 256, 0, stream>>>(x, agg, deg, Ws, Wn, bs, bn, out, Nn);
}